// SimilarityAttn_41781441856256
// MI455X (gfx1250) — compile-verified
//
#include <hip/hip_runtime.h>
#include <hip/hip_bf16.h>

// =====================================================================
// SimilarityAttn for MI455X (gfx1250, wave32, WMMA).
// All matmuls (QKV, Q*K^T, P*V, sim*sim^T, projections) run on the
// matrix pipe via V_WMMA_F32_16X16X4_F32 (fp32-exact; problem is
// compute-bound: ~101 GFLOP vs ~0.3 GB of HBM traffic).
// Tile staging uses GLOBAL_LOAD_ASYNC_TO_LDS_B128 (ASYNCcnt-tracked,
// no VGPR round-trip) with ping-pong LDS double buffering.
// =====================================================================

typedef float v2f __attribute__((ext_vector_type(2)));
typedef float v8f __attribute__((ext_vector_type(8)));

// 8-arg pattern per probe: (neg_a, A, neg_b, B, c_mod, C, reuse_a, reuse_b)
#define WMMA_F32(a, b, c) \
  __builtin_amdgcn_wmma_f32_16x16x4_f32(false, (a), false, (b), (short)0, (c), false, false)

#if defined(__AMDGCN__) && \
    __has_builtin(__builtin_amdgcn_global_load_async_to_lds_b128) && \
    __has_builtin(__builtin_amdgcn_s_wait_asynccnt)
#define HAVE_ASYNC_LDS 1
#else
#define HAVE_ASYNC_LDS 0
#endif

#if HAVE_ASYNC_LDS
// The builtin's params are pointers to vector_size(16) int in global (AS1)
// and LDS (AS3) address spaces (per hipcc's own diagnostic).
typedef int async_b128_t __attribute__((vector_size(16)));
typedef __attribute__((address_space(1))) async_b128_t* glb_b128_ptr;
typedef __attribute__((address_space(3))) async_b128_t* lds_b128_ptr;
#endif

// 16B-aligned global -> LDS copy of one float4 slot.
__device__ __forceinline__ void cp_g2l_b128(float* ldst, const float* gsrc) {
#if HAVE_ASYNC_LDS
  __builtin_amdgcn_global_load_async_to_lds_b128(
      (glb_b128_ptr)gsrc, (lds_b128_ptr)ldst, 0, 0);
#else
  *(float4*)ldst = *(const float4*)gsrc;
#endif
}
__device__ __forceinline__ void wait_async_lds() {
#if HAVE_ASYNC_LDS
  __builtin_amdgcn_s_wait_asynccnt(0);
#endif
}

constexpr int   Bsz   = 8, Ntok = 1032, Cdim = 768, NH = 12, HD = 64;
constexpr int   Mrows = Bsz * Ntok;            // 8256
constexpr int   NPREF = 8;
constexpr float ATTN_SCALE = 0.125f;           // 64^-0.5
constexpr float SIMSCALE   = 10.0f;
constexpr float PREFSCALE  = 2.0f;

__device__ __forceinline__ float rowmax16(float v) {
  #pragma unroll
  for (int o = 8; o > 0; o >>= 1) v = fmaxf(v, __shfl_xor(v, o, 16));
  return v;
}
__device__ __forceinline__ float rowsum16(float v) {
  #pragma unroll
  for (int o = 8; o > 0; o >>= 1) v += __shfl_xor(v, o, 16);
  return v;
}

// ---------------------------------------------------------------------
// GEMM: out[M,Nout] = A[M,K] @ W[Nout,K]^T + bias.
// 256 threads = 8 waves, wave grid 2x4, WG tile 32x64, K-block 32.
// Async double-buffered LDS staging (issue tile t+1, compute tile t).
// aHeadMajor: A is ctx [B,H,N,64] gathered as merged [B*N, 768].
// qkvScatter: write columns into q/k/v [3][B,H,N,64] instead of plain.
// ---------------------------------------------------------------------
__global__ __launch_bounds__(256) void gemm_wmma_kernel(
    const float* __restrict__ A, const float* __restrict__ W,
    const float* __restrict__ bias, float* __restrict__ outPlain,
    float* __restrict__ qkvScatter, int aHeadMajor)
{
  constexpr int KB = 32, KBP = 36;   // pad 36: 16B-aligned rows, no bank conflicts
  __shared__ float Xs[2][32][KBP];
  __shared__ float Wsh[2][64][KBP];

  const int tid  = threadIdx.x;
  const int wave = tid >> 5, lane = tid & 31;
  const int half = lane >> 4, l16 = lane & 15;
  const int waveM = wave >> 2, waveN = wave & 3;
  const int m0 = blockIdx.y * 32;
  const int n0 = blockIdx.x * 64;

  v8f acc;
  {
    const float bval = bias[n0 + waveN * 16 + l16];
    #pragma unroll
    for (int r = 0; r < 8; ++r) acc[r] = bval;
  }

  auto stage = [&](int k0, int buf) {
    {
      const int i = tid >> 3, j4 = (tid & 7) * 4;
      const int m = m0 + i;
      const float* src;
      if (aHeadMajor) {
        const int b = m / Ntok, n = m - b * Ntok, k = k0 + j4;
        src = A + (((size_t)(b * NH + (k >> 6)) * Ntok + n) * HD + (k & 63));
      } else {
        src = A + ((size_t)m * Cdim + k0 + j4);
      }
      cp_g2l_b128(&Xs[buf][i][j4], src);
      __builtin_prefetch(src + KB, 0, 0);   // global_prefetch_b8 (speculative)
    }
    #pragma unroll
    for (int t = 0; t < 2; ++t) {
      const int i = (tid >> 3) + t * 32, j4 = (tid & 7) * 4;
      const float* src = W + ((size_t)(n0 + i) * Cdim + k0 + j4);
      cp_g2l_b128(&Wsh[buf][i][j4], src);
      __builtin_prefetch(src + KB, 0, 0);
    }
  };

  stage(0, 0);
  constexpr int nblk = Cdim / KB;           // 24
  for (int blk = 0; blk < nblk; ++blk) {
    wait_async_lds();                        // this wave's staged tile complete
    __syncthreads();                         // publish + all done reading prev buf
    if (blk + 1 < nblk) stage((blk + 1) * KB, (blk + 1) & 1);
    const int buf = blk & 1;

    #pragma unroll
    for (int kk = 0; kk < KB / 4; ++kk) {
      v2f a, b;
      a.x = Xs[buf][waveM * 16 + l16][kk * 4 + 2 * half];
      a.y = Xs[buf][waveM * 16 + l16][kk * 4 + 2 * half + 1];
      b.x = Wsh[buf][waveN * 16 + l16][kk * 4 + 2 * half];
      b.y = Wsh[buf][waveN * 16 + l16][kk * 4 + 2 * half + 1];
      acc = WMMA_F32(a, b, acc);
    }
  }

  // --- epilogue: C/D layout row = r + 8*half, col = l16 ---
  #pragma unroll
  for (int r = 0; r < 8; ++r) {
    const int m    = m0 + waveM * 16 + r + 8 * half;
    const int nCol = n0 + waveN * 16 + l16;
    const float val = acc[r];
    if (qkvScatter) {
      const int which = nCol / Cdim;              // 0=q 1=k 2=v
      const int rem = nCol - which * Cdim;
      const int h = rem >> 6, d = rem & 63;
      const int b = m / Ntok, n = m - b * Ntok;
      qkvScatter[(size_t)which * ((size_t)Bsz * NH * Ntok * HD) +
                 (((size_t)(b * NH + h) * Ntok + n) * HD + d)] = val;
    } else {
      outPlain[(size_t)m * Cdim + nCol] = val;
    }
  }
}

// ---------------------------------------------------------------------
// Flash attention over [B,H,N,64] head-major q/k/v.
// 128 threads = 4 waves; each wave owns 16 query rows; key loop in
// blocks of 16, K/V tiles async-staged into ping-pong LDS buffers.
// Online softmax with half-wave shuffle reductions; P transposed to
// A-layout via per-wave LDS scratch for the P@V WMMAs.
// stage2: cosine-sim attention (scale 10, s<0 masked) + post-softmax
// prefix-column overwrite folded in as a correction WMMA pass.
// ---------------------------------------------------------------------
__global__ __launch_bounds__(128) void flash_attn_kernel(
    const float* Q, const float* Kt, const float* V,
    float* __restrict__ ctx, float* pref, int stage2)
{
  __shared__ float Ks[2][16][68];   // pad 68: 16B-aligned rows, conflict-free
  __shared__ float Vs[2][16][68];
  __shared__ float Ps[4][16][18];

  const int tid  = threadIdx.x;
  const int wave = tid >> 5, lane = tid & 31;
  const int half = lane >> 4, l16 = lane & 15;
  const int bh   = blockIdx.y;
  const int qr0  = blockIdx.x * 64 + wave * 16;
  const size_t base = (size_t)bh * Ntok * HD;

  // Q fragments for the whole hd=64 reduction (16 chunks of K=4)
  v2f qa[16];
  {
    int row = qr0 + l16; if (row >= Ntok) row = Ntok - 1;
    const float* qp = Q + base + (size_t)row * HD + 2 * half;
    #pragma unroll
    for (int c = 0; c < 16; ++c) { qa[c].x = qp[c * 4]; qa[c].y = qp[c * 4 + 1]; }
  }

  v8f acc[4] = {v8f{}, v8f{}, v8f{}, v8f{}};   // O tile 16x64
  float mrow[8], lrow[8], s0[8];
  #pragma unroll
  for (int r = 0; r < 8; ++r) { mrow[r] = -1e30f; lrow[r] = 0.f; s0[r] = -1e30f; }

  auto stageKV = [&](int kb, int buf) {
    #pragma unroll
    for (int it = 0; it < 2; ++it) {
      const int fidx = tid + it * 128;         // 256 float4 slots per tile
      const int rrow = fidx >> 4, col4 = (fidx & 15) * 4;
      int key = kb * 16 + rrow; if (key >= Ntok) key = Ntok - 1;
      cp_g2l_b128(&Ks[buf][rrow][col4], Kt + base + (size_t)key * HD + col4);
      cp_g2l_b128(&Vs[buf][rrow][col4], V  + base + (size_t)key * HD + col4);
    }
  };

  const int nKB = (Ntok + 15) / 16;            // 65 (last block half-valid)
  stageKV(0, 0);
  for (int kb = 0; kb < nKB; ++kb) {
    wait_async_lds();
    __syncthreads();
    if (kb + 1 < nKB) stageKV(kb + 1, (kb + 1) & 1);
    const int buf = kb & 1;

    // S = Q * K^T   (16 WMMAs, K staged in LDS)
    v8f sacc = {};
    #pragma unroll
    for (int c = 0; c < 16; ++c) {
      v2f kf;
      kf.x = Ks[buf][l16][c * 4 + 2 * half];
      kf.y = Ks[buf][l16][c * 4 + 2 * half + 1];
      sacc = WMMA_F32(qa[c], kf, sacc);
    }

    // online softmax (row = r + 8*half in C layout; stats via half-wave shfl)
    float p[8], fsc[8];
    #pragma unroll
    for (int r = 0; r < 8; ++r) {
      const int keycol = kb * 16 + l16;
      float s = stage2 ? sacc[r] * SIMSCALE : sacc[r] * ATTN_SCALE;
      const bool bad = (keycol >= Ntok) || (stage2 && s < 0.f);
      if (bad) s = -1e30f;
      if (kb == 0) s0[r] = s;                  // keep block-0 logits for prefix
      const float rm   = rowmax16(s);
      const float mnew = fmaxf(mrow[r], rm);
      const float f    = __expf(mrow[r] - mnew);
      const float pv   = (s < -1e29f) ? 0.f : __expf(s - mnew);
      lrow[r] = lrow[r] * f + rowsum16(pv);
      mrow[r] = mnew;
      p[r] = pv; fsc[r] = f;
    }
    #pragma unroll
    for (int r = 0; r < 8; ++r) {
      acc[0][r] *= fsc[r]; acc[1][r] *= fsc[r];
      acc[2][r] *= fsc[r]; acc[3][r] *= fsc[r];
    }

    // transpose P (C-layout -> A-layout) through per-wave LDS scratch
    #pragma unroll
    for (int r = 0; r < 8; ++r) Ps[wave][r + 8 * half][l16] = p[r];
    v2f pa[4];
    #pragma unroll
    for (int c = 0; c < 4; ++c) {
      pa[c].x = Ps[wave][l16][c * 4 + 2 * half];
      pa[c].y = Ps[wave][l16][c * 4 + 2 * half + 1];
    }

    // O += P * V   (16 WMMAs, V staged in LDS)
    #pragma unroll
    for (int ht = 0; ht < 4; ++ht) {
      #pragma unroll
      for (int c = 0; c < 4; ++c) {
        v2f vf;
        vf.x = Vs[buf][c * 4 + 2 * half][ht * 16 + l16];
        vf.y = Vs[buf][c * 4 + 2 * half + 1][ht * 16 + l16];
        acc[ht] = WMMA_F32(pa[c], vf, acc[ht]);
      }
    }
  }

  // normalize O by row sums
  float invl[8];
  #pragma unroll
  for (int r = 0; r < 8; ++r) invl[r] = 1.0f / lrow[r];
  #pragma unroll
  for (int ht = 0; ht < 4; ++ht)
    #pragma unroll
    for (int r = 0; r < 8; ++r) acc[ht][r] *= invl[r];

  if (!stage2) {
    // attn[:, :, :, :8] = exp(s0 - m_final) / Z_final
    #pragma unroll
    for (int r = 0; r < 8; ++r) {
      const int row = qr0 + r + 8 * half;
      const float aw = __expf(s0[r] - mrow[r]) * invl[r];
      if (l16 < NPREF && row < Ntok)
        pref[((size_t)bh * Ntok + row) * NPREF + l16] = aw;
    }
  } else {
    // post-softmax prefix overwrite: O += sum_{k<8} (2*attn_k - p_hat_k/Z) v_k
    float chat[8];
    #pragma unroll
    for (int r = 0; r < 8; ++r) {
      const int row = qr0 + r + 8 * half;
      float cc = 0.f;
      if (l16 < NPREF) {
        const float ph = (s0[r] < -1e29f) ? 0.f : __expf(s0[r] - mrow[r]) * invl[r];
        const float ap = (row < Ntok) ? pref[((size_t)bh * Ntok + row) * NPREF + l16] : 0.f;
        cc = PREFSCALE * ap - ph;
      }
      chat[r] = cc;
    }
    #pragma unroll
    for (int r = 0; r < 8; ++r) Ps[wave][r + 8 * half][l16] = chat[r];
    v2f pa[4];
    #pragma unroll
    for (int c = 0; c < 4; ++c) {
      pa[c].x = Ps[wave][l16][c * 4 + 2 * half];
      pa[c].y = Ps[wave][l16][c * 4 + 2 * half + 1];
    }
    #pragma unroll
    for (int ht = 0; ht < 4; ++ht) {
      #pragma unroll
      for (int c = 0; c < 4; ++c) {
        v2f vf;
        const float* vp = V + base + (size_t)(c * 4 + 2 * half) * HD + ht * 16 + l16;
        vf.x = vp[0]; vf.y = vp[HD];
        acc[ht] = WMMA_F32(pa[c], vf, acc[ht]);
      }
    }
  }

  // store ctx [B,H,N,64]
  #pragma unroll
  for (int r = 0; r < 8; ++r) {
    const int row = qr0 + r + 8 * half;
    if (row < Ntok) {
      float* cp = ctx + base + (size_t)row * HD + l16;
      cp[0]  = acc[0][r]; cp[16] = acc[1][r];
      cp[32] = acc[2][r]; cp[48] = acc[3][r];
    }
  }
}

// ---------------------------------------------------------------------
// Per-(b,n,h) L2 normalization: out1 [B,N,C] -> sim_tok [B,H,N,64].
// One wave per token-head; wave-wide shuffle reduction.
// ---------------------------------------------------------------------
__global__ __launch_bounds__(256) void normalize_heads_kernel(
    const float* __restrict__ out1, float* __restrict__ simTok)
{
  const int gw = blockIdx.x * 8 + (threadIdx.x >> 5);     // = bh*Ntok + n
  const int lane = threadIdx.x & 31;
  if (gw >= Bsz * NH * Ntok) return;
  const int bh = gw / Ntok, n = gw - bh * Ntok;
  const int b = bh / NH, h = bh - b * NH;
  const float* src = out1 + (size_t)(b * Ntok + n) * Cdim + h * HD;
  const float v0 = src[lane], v1 = src[lane + 32];
  float ss = v0 * v0 + v1 * v1;
  #pragma unroll
  for (int o = 16; o > 0; o >>= 1) ss += __shfl_xor(ss, o, 32);
  const float inv = 1.0f / fmaxf(sqrtf(ss), 1e-12f);
  float* dst = simTok + (size_t)gw * HD;
  dst[lane] = v0 * inv; dst[lane + 32] = v1 * inv;
}

// ---------------------------------------------------------------------
extern "C" void kernel_launch(void* const* d_in, const int* in_sizes, int n_in,
                              void* d_out, int out_size, void* d_ws, size_t ws_size,
                              hipStream_t stream)
{
  const float* x     = (const float*)d_in[0];
  const float* Wqkv  = (const float*)d_in[1];
  const float* bqkv  = (const float*)d_in[2];
  const float* Wproj = (const float*)d_in[3];
  const float* bproj = (const float*)d_in[4];
  float* out = (float*)d_out;

  const size_t SZ = (size_t)Bsz * NH * Ntok * HD;   // 6,340,608 floats
  float* ws   = (float*)d_ws;
  float* q    = ws;                // q / later sim_tok
  float* k    = ws + SZ;           // k / later out1
  float* v    = ws + 2 * SZ;       // v (live through stage 2)
  float* ctx  = ws + 3 * SZ;       // ctx1 / later ctx2
  float* pref = ws + 4 * SZ;       // attn prefix [B,H,N,8]
  float* out1 = k;
  float* simt = q;

  // 1. qkv = x @ Wqkv^T + bqkv, scattered into head-major q/k/v
  gemm_wmma_kernel<<<dim3(3 * Cdim / 64, Mrows / 32), 256, 0, stream>>>(
      x, Wqkv, bqkv, nullptr, q, 0);
  // 2. stage-1 softmax attention (writes ctx1 and attn prefix columns)
  flash_attn_kernel<<<dim3((Ntok + 63) / 64, Bsz * NH), 128, 0, stream>>>(
      q, k, v, ctx, pref, 0);
  // 3. out1 = merge(ctx1) @ Wproj^T + bproj
  gemm_wmma_kernel<<<dim3(Cdim / 64, Mrows / 32), 256, 0, stream>>>(
      ctx, Wproj, bproj, out1, nullptr, 1);
  // 4. sim_tok = normalize(split_heads(out1))
  normalize_heads_kernel<<<dim3(Bsz * NH * Ntok / 8), 256, 0, stream>>>(out1, simt);
  // 5. stage-2 cosine-sim attention with prefix overwrite (writes ctx2)
  flash_attn_kernel<<<dim3((Ntok + 63) / 64, Bsz * NH), 128, 0, stream>>>(
      simt, simt, v, ctx, pref, 1);
  // 6. out = merge(ctx2) @ Wproj^T + bproj
  gemm_wmma_kernel<<<dim3(Cdim / 64, Mrows / 32), 256, 0, stream>>>(
      ctx, Wproj, bproj, out, nullptr, 1);
}